// Attn_44710609551743
// MI455X (gfx1250) — compile-verified
//
#include <hip/hip_runtime.h>
#include <hip/hip_bf16.h>
#include <math.h>

typedef float v2f __attribute__((ext_vector_type(2)));
typedef float v4f __attribute__((ext_vector_type(4)));
typedef float v8f __attribute__((ext_vector_type(8)));

#define HIDDEN 512
#define SEQ    4096
#define BATCH  64
#define SCHUNK 64   // s values per wave in the streaming pass

// ---------------------------------------------------------------------------
// Kernel 1: v = hidden(64x512) @ W(512x512)   [W stored row-major as W[k][h]]
// One wave per 16x16 output tile, V_WMMA_F32_16X16X4_F32 over K=512 (128 steps).
// ---------------------------------------------------------------------------
__global__ __launch_bounds__(32) void compute_v_wmma(
    const float* __restrict__ hidden,  // [64,512]
    const float* __restrict__ W,       // [512,512]  (k-major)
    float* __restrict__ v)             // [64,512]
{
  const int lane  = threadIdx.x;       // 0..31 (wave32)
  const int tn    = blockIdx.x * 16;   // N tile offset (h)
  const int tm    = blockIdx.y * 16;   // M tile offset (b)
  const int mrow  = lane & 15;         // M (for A) / N (for B, C/D)
  const int koff  = (lane >> 4) * 2;   // lanes 16-31 hold K+2,K+3

  // induction-variable pointers: offsets fold into load immediates
  const float* ap = hidden + (tm + mrow) * HIDDEN + koff;
  const float* bp = W + koff * HIDDEN + tn + mrow;

  v8f acc = {};
  for (int k = 0; k < HIDDEN; k += 4) {
    v2f a, b;
    a.x = ap[0];
    a.y = ap[1];
    b.x = bp[0];
    b.y = bp[HIDDEN];
    acc = __builtin_amdgcn_wmma_f32_16x16x4_f32(
        /*neg_a=*/false, a, /*neg_b=*/false, b,
        /*c_mod=*/(short)0, acc, /*reuse_a=*/false, /*reuse_b=*/false);
    ap += 4;
    bp += 4 * HIDDEN;
  }

  // C/D layout: VGPR r -> M = r + 8*(lane/16), N = lane%16
  const int mbase = (lane >> 4) * 8;
#pragma unroll
  for (int r = 0; r < 8; ++r)
    v[(tm + mbase + r) * HIDDEN + tn + mrow] = acc[r];
}

// ---------------------------------------------------------------------------
// Kernel 2: c[b] = <bias, hidden[b,:]>   (tiny)
// ---------------------------------------------------------------------------
__global__ __launch_bounds__(64) void compute_c(
    const float* __restrict__ hidden, const float* __restrict__ bias,
    float* __restrict__ c)
{
  const int b = threadIdx.x;  // 64 threads
  float s = 0.f;
  for (int k = 0; k < HIDDEN; ++k)
    s = fmaf(bias[k], hidden[b * HIDDEN + k], s);
  c[b] = s;
}

// ---------------------------------------------------------------------------
// Kernel 3: energies[b][s] = <E[s,b,:], v[b,:]> + c[b]
// Streams 512 MB of encoder_outputs once (NT loads); v chunk lives in VGPRs.
// One wave per (b, SCHUNK-wide s-chunk): 64 * 64 = 4096 waves.
// ---------------------------------------------------------------------------
__global__ __launch_bounds__(256) void energies_kernel(
    const float* __restrict__ E,   // [4096,64,512]
    const float* __restrict__ v,   // [64,512]
    const float* __restrict__ c,   // [64]
    float* __restrict__ energ)     // [64,4096]  (b-major)
{
  const int gwave = (int)((blockIdx.x * blockDim.x + threadIdx.x) >> 5);
  const int lane  = threadIdx.x & 31;
  const int b     = gwave / (SEQ / SCHUNK);            // 0..63
  const int s0    = (gwave % (SEQ / SCHUNK)) * SCHUNK; // s-chunk base

  // lane l caches v[b, 16l .. 16l+15] in registers (regular loads -> L2 hot)
  const v4f* vp = (const v4f*)(v + (size_t)b * HIDDEN + lane * 16);
  const v4f v0 = vp[0], v1 = vp[1], v2 = vp[2], v3 = vp[3];
  const float cb = c[b];

  const float* ebase = E + ((size_t)s0 * BATCH + b) * HIDDEN + lane * 16;

  for (int si = 0; si < SCHUNK; ++si) {
    const v4f* ep = (const v4f*)(ebase + (size_t)si * BATCH * HIDDEN);
    // non-temporal: single-use stream, keep L2 for v
    v4f e0 = __builtin_nontemporal_load(ep + 0);
    v4f e1 = __builtin_nontemporal_load(ep + 1);
    v4f e2 = __builtin_nontemporal_load(ep + 2);
    v4f e3 = __builtin_nontemporal_load(ep + 3);

    float acc = 0.f;
    acc = fmaf(e0.x, v0.x, acc); acc = fmaf(e0.y, v0.y, acc);
    acc = fmaf(e0.z, v0.z, acc); acc = fmaf(e0.w, v0.w, acc);
    acc = fmaf(e1.x, v1.x, acc); acc = fmaf(e1.y, v1.y, acc);
    acc = fmaf(e1.z, v1.z, acc); acc = fmaf(e1.w, v1.w, acc);
    acc = fmaf(e2.x, v2.x, acc); acc = fmaf(e2.y, v2.y, acc);
    acc = fmaf(e2.z, v2.z, acc); acc = fmaf(e2.w, v2.w, acc);
    acc = fmaf(e3.x, v3.x, acc); acc = fmaf(e3.y, v3.y, acc);
    acc = fmaf(e3.z, v3.z, acc); acc = fmaf(e3.w, v3.w, acc);

    // wave32 reduction
#pragma unroll
    for (int off = 16; off >= 1; off >>= 1)
      acc += __shfl_xor(acc, off, 32);

    if (lane == 0) energ[(size_t)b * SEQ + s0 + si] = acc + cb;
  }
}

// ---------------------------------------------------------------------------
// Kernel 4: softmax over s (4096) per b; writes out[b,0,s] directly.
// ---------------------------------------------------------------------------
__global__ __launch_bounds__(256) void softmax_kernel(
    const float* __restrict__ energ,  // [64,4096]
    float* __restrict__ out)          // [64,1,4096]
{
  const int b = blockIdx.x;
  const int t = threadIdx.x;  // 256
  __shared__ float red[256];

  const float* row = energ + (size_t)b * SEQ;
  float m = -INFINITY;
  for (int s = t; s < SEQ; s += 256) m = fmaxf(m, row[s]);
  red[t] = m; __syncthreads();
#pragma unroll
  for (int off = 128; off >= 1; off >>= 1) {
    if (t < off) red[t] = fmaxf(red[t], red[t + off]);
    __syncthreads();
  }
  m = red[0];
  __syncthreads();

  float sum = 0.f;
  float* orow = out + (size_t)b * SEQ;
  for (int s = t; s < SEQ; s += 256) {
    float e = __expf(row[s] - m);
    orow[s] = e;
    sum += e;
  }
  red[t] = sum; __syncthreads();
#pragma unroll
  for (int off = 128; off >= 1; off >>= 1) {
    if (t < off) red[t] += red[t + off];
    __syncthreads();
  }
  const float inv = 1.0f / red[0];
  for (int s = t; s < SEQ; s += 256) orow[s] *= inv;
}

// ---------------------------------------------------------------------------
extern "C" void kernel_launch(void* const* d_in, const int* in_sizes, int n_in,
                              void* d_out, int out_size, void* d_ws,
                              size_t ws_size, hipStream_t stream) {
  (void)in_sizes; (void)n_in; (void)out_size; (void)ws_size;

  const float* hidden = (const float*)d_in[0];  // [64,512]
  const float* E      = (const float*)d_in[1];  // [4096,64,512]
  const float* W      = (const float*)d_in[2];  // [512,512]
  const float* bias   = (const float*)d_in[3];  // [512]
  float* out          = (float*)d_out;          // [64,1,4096]

  // workspace carve-up
  float* ws    = (float*)d_ws;
  float* v     = ws;                         // 64*512
  float* cvec  = v + BATCH * HIDDEN;         // 64
  float* energ = cvec + BATCH;               // 64*4096

  // 1) v = hidden @ W  (WMMA f32 16x16x4), 32 N-tiles x 4 M-tiles
  compute_v_wmma<<<dim3(HIDDEN / 16, BATCH / 16), 32, 0, stream>>>(hidden, W, v);
  // 2) c[b]
  compute_c<<<1, 64, 0, stream>>>(hidden, bias, cvec);
  // 3) energies: 4096 waves = 512 blocks x 256 threads
  energies_kernel<<<(BATCH * (SEQ / SCHUNK) * 32) / 256, 256, 0, stream>>>(
      E, v, cvec, energ);
  // 4) softmax over s, writes out[b,0,s] in energ's layout
  softmax_kernel<<<BATCH, 256, 0, stream>>>(energ, out);
}